// MultiBoxLoss_5334349381819
// MI455X (gfx1250) — compile-verified
//
#include <hip/hip_runtime.h>
#include <hip/hip_bf16.h>

// MultiBoxLoss for MI455X (gfx1250, wave32).
// Grid: 128 workgroups (one per (b,f)), 256 threads (8 wave32) each.
// Memory-bound: ~210 MB streamed once -> ~9us floor at 23.3 TB/s.
// WMMA f32_16x16x4 used as an exact f32 cross-lane sum engine (B = ones).

namespace {
constexpr int kP = 16384;   // priors
constexpr int kO = 10;      // objects (truths) per image
constexpr int kC = 21;      // classes
constexpr int kG = 128;     // B*F = 8*16 groups
constexpr int kT = 256;     // threads per block (8 wave32)
}

typedef __attribute__((ext_vector_type(2))) float v2f;
typedef __attribute__((ext_vector_type(8))) float v8f;

// Exact f32 sum over the 32 lanes of a wave using V_WMMA_F32_16X16X4_F32.
// A(16x4): lane m<16 -> A[m][0]=x_m, A[m][1]=0 ; lane m+16 -> A[m][2]=x_{m+16}, A[m][3]=0.
// B(4x16) = ones (layout-invariant). D[m][n] = x_m + x_{m+16}.
// Per-lane sum of D's 8 VGPRs gives half-sums; shfl_xor(16) completes the wave sum.
// Must be called in uniform control flow (EXEC all ones).
__device__ __forceinline__ float wave_sum_wmma(float x) {
  v2f a; a[0] = x;    a[1] = 0.0f;
  v2f b; b[0] = 1.0f; b[1] = 1.0f;
  v8f c = {0.f, 0.f, 0.f, 0.f, 0.f, 0.f, 0.f, 0.f};
  v8f d = __builtin_amdgcn_wmma_f32_16x16x4_f32(
      /*neg_a=*/false, a, /*neg_b=*/false, b,
      /*c_mod=*/(short)0, c, /*reuse_a=*/false, /*reuse_b=*/false);
  float t = d[0] + d[1] + d[2] + d[3] + d[4] + d[5] + d[6] + d[7];
  t += __shfl_xor(t, 16, 32);
  return t;  // uniform across the wave
}

__device__ __forceinline__ float smoothl1(float d) {
  float a = fabsf(d);
  return (a < 1.0f) ? 0.5f * d * d : a - 0.5f;
}

__global__ void mbl_init(float* acc) {
  acc[0] = 0.0f;                 // sum loss_l
  acc[1] = 0.0f;                 // sum loss_c
  ((unsigned*)acc)[2] = 0u;      // sum num_pos
}

__global__ __launch_bounds__(kT)
void mbl_main(const float* __restrict__ loc,     // (8,16,P,4)
              const float* __restrict__ pc,      // (8,16,P,21)
              const float* __restrict__ tgt,     // (16,8,10,6)
              const float* __restrict__ priors,  // (P,4) cx,cy,w,h
              float* __restrict__ acc,
              float* __restrict__ mined_ws) {    // (128,P) scratch (L2-resident)
  const int g = blockIdx.x;           // g = b*16 + f
  const int b = g >> 4, f = g & 15;
  const int tid = threadIdx.x;
  const int lane = tid & 31;

  __shared__ float s_x1[kO], s_y1[kO], s_x2[kO], s_y2[kO], s_lab[kO], s_ar[kO];
  __shared__ unsigned long long s_bpk[kO];   // per-truth best-prior key (ds_max_u64)
  __shared__ unsigned s_hist[kT];            // 256-bucket radix histogram
  __shared__ float s_lossl, s_losscp, s_sumgt, s_negc;
  __shared__ unsigned s_np, s_cnt, s_kk, s_pref;

  if (tid < kO) {
    const float* tr = tgt + ((size_t)(f * 8 + b) * kO + tid) * 6;
    float x1 = tr[0], y1 = tr[1], x2 = tr[2], y2 = tr[3];
    s_x1[tid] = x1; s_y1[tid] = y1; s_x2[tid] = x2; s_y2[tid] = y2;
    s_lab[tid] = tr[4];
    s_ar[tid] = (x2 - x1) * (y2 - y1);
    s_bpk[tid] = 0ull;
  }
  if (tid == 0) { s_lossl = 0.f; s_losscp = 0.f; s_sumgt = 0.f; s_negc = 0.f;
                  s_np = 0u; s_cnt = 0u; }
  __syncthreads();

  const float4* pri4 = (const float4*)priors;  // priors: 256 KB, L2-resident

  // ---- Phase 1: per-truth argmax over priors (first-index tie-break) ----
  unsigned long long bk[kO];
#pragma unroll
  for (int t = 0; t < kO; ++t) bk[t] = 0ull;

  for (int p = tid; p < kP; p += kT) {
    float4 pr = pri4[p];
    float px1 = pr.x - 0.5f * pr.z, py1 = pr.y - 0.5f * pr.w;
    float px2 = pr.x + 0.5f * pr.z, py2 = pr.y + 0.5f * pr.w;
    float pa = pr.z * pr.w;
#pragma unroll
    for (int t = 0; t < kO; ++t) {
      float ix = fmaxf(fminf(s_x2[t], px2) - fmaxf(s_x1[t], px1), 0.f);
      float iy = fmaxf(fminf(s_y2[t], py2) - fmaxf(s_y1[t], py1), 0.f);
      float inter = ix * iy;
      float iou = inter / (s_ar[t] + pa - inter);
      // iou >= 0 so float bits are order-preserving; ~p breaks ties toward low p
      unsigned long long key = ((unsigned long long)__float_as_uint(iou) << 32)
                             | (unsigned long long)(0xFFFFFFFFu - (unsigned)p);
      bk[t] = (key > bk[t]) ? key : bk[t];
    }
  }
#pragma unroll
  for (int t = 0; t < kO; ++t) atomicMax(&s_bpk[t], bk[t]);  // ds_max_u64
  __syncthreads();

  unsigned bpi[kO];
#pragma unroll
  for (int t = 0; t < kO; ++t)
    bpi[t] = 0xFFFFFFFFu - (unsigned)(s_bpk[t] & 0xFFFFFFFFull);

  // ---- Phase 2: per-prior match + CE + SmoothL1; stream mined to scratch ----
  const size_t gp = (size_t)g * kP;
  const float* locg = loc + gp * 4;
  const float* pcg  = pc  + gp * kC;
  float* mg = mined_ws + gp;

  float tl = 0.f, tcp = 0.f; unsigned tnp = 0u;
  for (int p = tid; p < kP; p += kT) {
    float4 pr = pri4[p];
    float px1 = pr.x - 0.5f * pr.z, py1 = pr.y - 0.5f * pr.w;
    float px2 = pr.x + 0.5f * pr.z, py2 = pr.y + 0.5f * pr.w;
    float pa = pr.z * pr.w;

    float bo = -1.0f; int bi = 0;
#pragma unroll
    for (int t = 0; t < kO; ++t) {
      float ix = fmaxf(fminf(s_x2[t], px2) - fmaxf(s_x1[t], px1), 0.f);
      float iy = fmaxf(fminf(s_y2[t], py2) - fmaxf(s_y1[t], py1), 0.f);
      float inter = ix * iy;
      float iou = inter / (s_ar[t] + pa - inter);
      if (iou > bo) { bo = iou; bi = t; }     // argmax axis=0, first occurrence
    }
#pragma unroll
    for (int t = 0; t < kO; ++t)
      if ((unsigned)p == bpi[t]) { bo = 2.0f; bi = t; }  // forced positives, last t wins

    int conf = (bo < 0.5f) ? 0 : ((int)s_lab[bi] + 1);

    // logsumexp over 21 classes (the dominant HBM stream)
    const float* row = pcg + (size_t)p * kC;
    __builtin_prefetch(row + (size_t)kC * kT, 0, 0);   // global_prefetch_b8
    float r[kC];
#pragma unroll
    for (int c2 = 0; c2 < kC; ++c2) r[c2] = row[c2];
    float mx = r[0];
#pragma unroll
    for (int c2 = 1; c2 < kC; ++c2) mx = fmaxf(mx, r[c2]);
    float se = 0.f;
#pragma unroll
    for (int c2 = 0; c2 < kC; ++c2) se += __expf(r[c2] - mx);
    float ce = mx + __logf(se) - r[conf];

    float mined = ce;
    if (conf > 0) {
      tnp++; tcp += ce;
      float gx = ((s_x1[bi] + s_x2[bi]) * 0.5f - pr.x) / (0.1f * pr.z);
      float gy = ((s_y1[bi] + s_y2[bi]) * 0.5f - pr.y) / (0.1f * pr.w);
      float gw = __logf((s_x2[bi] - s_x1[bi]) / pr.z) * 5.0f;   // 1/0.2
      float gh = __logf((s_y2[bi] - s_y1[bi]) / pr.w) * 5.0f;
      const float* lr = locg + (size_t)p * 4;
      tl += smoothl1(lr[0] - gx) + smoothl1(lr[1] - gy)
          + smoothl1(lr[2] - gw) + smoothl1(lr[3] - gh);
      mined = 0.f;
    }
    mg[p] = mined;
  }

  // Block reduction: WMMA wave sums -> LDS atomics
  float wl = wave_sum_wmma(tl);
  float wc = wave_sum_wmma(tcp);
  if (lane == 0) { atomicAdd(&s_lossl, wl); atomicAdd(&s_losscp, wc); }
  atomicAdd(&s_np, tnp);
  __threadfence();          // make mined stores visible for re-read
  __syncthreads();

  // ---- Phase 3: hard-negative mining. Top-k sum via MSD radix select ----
  const int np = (int)s_np;
  int k = 3 * np; if (k > kP - 1) k = kP - 1;
  if (k > 0) {
    if (tid == 0) { s_kk = (unsigned)k; s_pref = 0u; }
    for (int rnd = 0; rnd < 4; ++rnd) {
      const int shift = 24 - 8 * rnd;
      const unsigned hm = (rnd == 0) ? 0u : (0xFFFFFFFFu << (32 - 8 * rnd));
      s_hist[tid] = 0u;
      __syncthreads();
      const unsigned pref = s_pref;
      for (int p = tid; p < kP; p += kT) {
        unsigned v = __float_as_uint(mg[p]);           // mined >= 0: bit order == value order
        if ((v & hm) == pref) atomicAdd(&s_hist[(v >> shift) & 255u], 1u);
      }
      __syncthreads();
      if (tid == 0) {
        unsigned kk = s_kk, pf = s_pref;
        for (int bkt = 255; bkt >= 0; --bkt) {
          unsigned c = s_hist[bkt];
          if (kk <= c) { pf |= ((unsigned)bkt) << shift; break; }
          kk -= c;
        }
        s_kk = kk; s_pref = pf;
      }
      __syncthreads();
    }
    const unsigned T = s_pref;          // bits of the k-th largest mined value
    const float Tf = __uint_as_float(T);
    float ts = 0.f; unsigned tc = 0u;
    for (int p = tid; p < kP; p += kT) {
      float v = mg[p];
      if (__float_as_uint(v) > T) { ts += v; tc++; }
    }
    float wsum = wave_sum_wmma(ts);     // uniform branch (k block-uniform): EXEC all ones
    if (lane == 0) atomicAdd(&s_sumgt, wsum);
    atomicAdd(&s_cnt, tc);
    __syncthreads();
    if (tid == 0)
      s_negc = s_sumgt + (float)(k - (int)s_cnt) * Tf;   // ties at T handled exactly
  }
  __syncthreads();

  if (tid == 0) {
    atomicAdd(&acc[0], s_lossl);
    atomicAdd(&acc[1], s_losscp + s_negc);
    atomicAdd((unsigned*)acc + 2, (unsigned)np);
  }
}

__global__ void mbl_final(const float* __restrict__ acc, float* __restrict__ out) {
  float N = (float)(((const unsigned*)acc)[2]) * 128.0f;  // sum(num_pos) * F * B
  out[0] = acc[0] / N;
  out[1] = acc[1] / N;
}

extern "C" void kernel_launch(void* const* d_in, const int* in_sizes, int n_in,
                              void* d_out, int out_size, void* d_ws, size_t ws_size,
                              hipStream_t stream) {
  (void)in_sizes; (void)n_in; (void)out_size; (void)ws_size;
  const float* loc = (const float*)d_in[0];   // (8,16,16384,4)
  const float* pc  = (const float*)d_in[1];   // (8,16,16384,21)
  const float* tgt = (const float*)d_in[2];   // (16,8,10,6)
  const float* pri = (const float*)d_in[3];   // (16384,4)
  float* acc   = (float*)d_ws;                // [0]=loss_l [1]=loss_c [2]=num_pos(uint)
  float* mined = (float*)d_ws + 16;           // 128*16384 floats (8 MB), L2-resident

  hipLaunchKernelGGL(mbl_init,  dim3(1),  dim3(1),  0, stream, acc);
  hipLaunchKernelGGL(mbl_main,  dim3(kG), dim3(kT), 0, stream,
                     loc, pc, tgt, pri, acc, mined);
  hipLaunchKernelGGL(mbl_final, dim3(1),  dim3(1),  0, stream, acc, (float*)d_out);
}